// LengthRegulator_29429115912691
// MI455X (gfx1250) — compile-verified
//
#include <hip/hip_runtime.h>

#define NB   16      // batch
#define TP   512     // phone length
#define CH   256     // channels (D == F)
#define TMEL 2560    // mel_max_length
#define LN_EPS 1e-5f

#define XS_STRIDE 260   // staging row stride (floats): 16B-aligned rows, conflict-free b64 reads
#define HS_STRIDE 257   // layernorm scratch stride

typedef __attribute__((ext_vector_type(2))) float v2f;
typedef __attribute__((ext_vector_type(8))) float v8f;

// ---------------------------------------------------------------------------
// Weight transpose into WMMA-B-native layout:
//   wt[kk][cp][2*f + j] = w[f][2*cp + j][kk],  kk<3, cp<128, f<256, j<2
// so a B fragment (K=c..c+3 x N=f0..f0+15) is one coalesced b64 load per lane.
// ---------------------------------------------------------------------------
__global__ __launch_bounds__(256) void transpose_w_kernel(
    const float* __restrict__ w, float* __restrict__ wt)
{
  const int e = blockIdx.x * 256 + threadIdx.x;    // < 3*128*512 = 196608
  const int kk = e >> 16;          // e / 65536
  const int r  = e & 65535;
  const int cp = r >> 9;
  const int q  = r & 511;
  const int f  = q >> 1;
  const int j  = q & 1;
  wt[e] = w[(f * CH + (2 * cp + j)) * 3 + kk];
}

// ---------------------------------------------------------------------------
// Fused Conv1d(K=3, same) + bias + ReLU + LayerNorm.
// Grid: (TP/64, NB), block 256 threads = 8 waves.
// Bulk x staging via GLOBAL_LOAD_ASYNC_TO_LDS_B128 (ASYNCcnt); halo rows via
// regular predicated loads. WMMA: V_WMMA_F32_16X16X4_F32, fp32 precision.
// ---------------------------------------------------------------------------
__global__ __launch_bounds__(256) void conv_relu_ln_kernel(
    const float* __restrict__ in, const float* __restrict__ wt,
    const float* __restrict__ cbias, const float* __restrict__ gamma,
    const float* __restrict__ beta, float* __restrict__ out)
{
  __shared__ float smem[66 * XS_STRIDE];   // staging 66 rows; reused as hs[64][HS_STRIDE]
  __shared__ float s_mean[64], s_rstd[64];

  const int tid = threadIdx.x;
  const int b  = blockIdx.y;
  const int t0 = blockIdx.x * 64;

  // --- bulk staging: rows 1..64 (t = t0..t0+63, always in range), async to LDS
  {
    const float* src = in + ((size_t)b * TP + t0) * CH;
    #pragma unroll 4
    for (int i = tid; i < 64 * 64; i += 256) {      // 4096 float4 tiles, 16/thread
      const int r = i >> 6, c4 = (i & 63) * 4;
      const float* gp = src + r * CH + c4;
      const unsigned lp =
          (unsigned)(uintptr_t)(&smem[(1 + r) * XS_STRIDE + c4]);
      asm volatile("global_load_async_to_lds_b128 %0, %1, off"
                   :: "v"(lp), "v"(gp) : "memory");
    }
  }
  // --- halo rows 0 and 65 (zero-padded at batch edges)
  {
    const int t_top = t0 - 1, t_bot = t0 + 64;
    smem[0 * XS_STRIDE + tid] =
        (t_top >= 0) ? in[((size_t)b * TP + t_top) * CH + tid] : 0.f;
    smem[65 * XS_STRIDE + tid] =
        (t_bot < TP) ? in[((size_t)b * TP + t_bot) * CH + tid] : 0.f;
  }
  asm volatile("s_wait_asynccnt 0x0" ::: "memory");
  __syncthreads();

  const int wave = tid >> 5, lane = tid & 31;
  const int half = lane >> 4, m = lane & 15;
  const int tw = wave & 3;            // t-tile within block (16 rows each)
  const int fb = (wave >> 2) * 128;   // feature base: 0 or 128

  v8f acc[8] = {};

  // Conv as 3 shifted GEMMs, K = 256 each, stepped by 4 for 16x16x4 WMMA.
  for (int kk = 0; kk < 3; ++kk) {
    const int arow = tw * 16 + m + kk;                 // LDS row = global t - (t0-1)
    const float* wkk = wt + (size_t)kk * 128 * 512;
    for (int c0 = 0; c0 < CH; c0 += 4) {
      const int c = c0 + 2 * half;                     // A: half 0 -> K 0,1 ; half 1 -> K 2,3
      v2f a;
      a.x = smem[arow * XS_STRIDE + c];
      a.y = smem[arow * XS_STRIDE + c + 1];
      const float* wrow = wkk + ((c0 >> 1) + half) * 512;  // B: same half-wave K split
      #pragma unroll
      for (int nt = 0; nt < 8; ++nt) {
        const int f = fb + nt * 16 + m;
        const v2f bf = *(const v2f*)(wrow + 2 * f);    // one coalesced b64 per lane
        acc[nt] = __builtin_amdgcn_wmma_f32_16x16x4_f32(
            false, a, false, bf, (short)0, acc[nt], false, false);
      }
    }
  }
  __syncthreads();   // staging reads done -> reuse smem as hs[64][HS_STRIDE]

  // bias + ReLU, spill tile to LDS for the cross-feature LayerNorm.
  // C/D layout: VGPR r holds row r (lanes 0-15) and row r+8 (lanes 16-31).
  const int rbase = tw * 16 + half * 8;
  #pragma unroll
  for (int nt = 0; nt < 8; ++nt) {
    const int col = fb + nt * 16 + m;
    const float bv = cbias[col];
    #pragma unroll
    for (int r = 0; r < 8; ++r) {
      smem[(rbase + r) * HS_STRIDE + col] = fmaxf(acc[nt][r] + bv, 0.f);
    }
  }
  __syncthreads();

  // LayerNorm: 4 threads per row (64 cols each), wave32 shuffle reduction.
  {
    const int row = tid >> 2, sub = tid & 3;
    const int cb0 = sub * 64;
    float s = 0.f, ss = 0.f;
    for (int c = cb0; c < cb0 + 64; ++c) {
      const float vv = smem[row * HS_STRIDE + c];
      s += vv; ss += vv * vv;
    }
    s += __shfl_xor(s, 1); ss += __shfl_xor(ss, 1);
    s += __shfl_xor(s, 2); ss += __shfl_xor(ss, 2);
    if (sub == 0) {
      const float mean = s * (1.f / 256.f);
      const float var  = ss * (1.f / 256.f) - mean * mean;  // biased var (jnp.var)
      s_mean[row] = mean;
      s_rstd[row] = rsqrtf(var + LN_EPS);
    }
  }
  __syncthreads();
  {
    const int row = tid >> 2, sub = tid & 3;
    const float mean = s_mean[row], rstd = s_rstd[row];
    float* op = out + ((size_t)(b * TP + t0 + row)) * CH;
    const int cb0 = sub * 64;
    for (int c = cb0; c < cb0 + 64; ++c) {
      op[c] = (smem[row * HS_STRIDE + c] - mean) * rstd * gamma[c] + beta[c];
    }
  }
}

// ---------------------------------------------------------------------------
// Linear 256->1 + ReLU. One wave per row, wave32 shuffle reduction.
// ---------------------------------------------------------------------------
__global__ __launch_bounds__(256) void linear_relu_kernel(
    const float* __restrict__ h, const float* __restrict__ lw,
    const float* __restrict__ lb, float* __restrict__ dur)
{
  const int wave = threadIdx.x >> 5, lane = threadIdx.x & 31;
  const int row = blockIdx.x * 8 + wave;
  const float4* hp = (const float4*)(h + (size_t)row * CH);
  const float4* wp = (const float4*)lw;
  const float4 a0 = hp[lane * 2], a1 = hp[lane * 2 + 1];
  const float4 w0 = wp[lane * 2], w1 = wp[lane * 2 + 1];
  float d = a0.x * w0.x + a0.y * w0.y + a0.z * w0.z + a0.w * w0.w
          + a1.x * w1.x + a1.y * w1.y + a1.z * w1.z + a1.w * w1.w;
  #pragma unroll
  for (int o = 16; o >= 1; o >>= 1) d += __shfl_xor(d, o);
  if (lane == 0) dur[row] = fmaxf(d + lb[0], 0.f);
}

// ---------------------------------------------------------------------------
// Per-batch inclusive scan of durations (Hillis-Steele in LDS).
// ---------------------------------------------------------------------------
__global__ __launch_bounds__(512) void cumsum_kernel(
    const int* __restrict__ tgt, int* __restrict__ ends)
{
  __shared__ int sc[512];
  const int b = blockIdx.x, tid = threadIdx.x;
  sc[tid] = tgt[b * TP + tid];
  __syncthreads();
  for (int off = 1; off < 512; off <<= 1) {
    const int v = (tid >= off) ? sc[tid - off] : 0;
    __syncthreads();
    sc[tid] += v;
    __syncthreads();
  }
  ends[b * TP + tid] = sc[tid];
}

// ---------------------------------------------------------------------------
// Length regulator: out[b,t,:] = x[b, j, :] where ends[j-1] <= t < ends[j],
// zeros past total duration. One wave per mel frame; binary search on LDS
// copy of ends; float4 stores -> coalesced 40 MB write stream (the bound).
// ---------------------------------------------------------------------------
__global__ __launch_bounds__(256) void regulate_kernel(
    const float* __restrict__ x, const int* __restrict__ ends,
    float* __restrict__ out)
{
  __shared__ int se[512];
  const int tid = threadIdx.x;
  const int b = blockIdx.y;
  se[tid]       = ends[b * TP + tid];
  se[tid + 256] = ends[b * TP + tid + 256];
  __syncthreads();

  const int rw = tid >> 5, lane = tid & 31;
  const int t = blockIdx.x * 8 + rw;
  float4 v0 = make_float4(0.f, 0.f, 0.f, 0.f), v1 = v0;
  if (t < se[511]) {
    int lo = 0, hi = 511;
    while (lo < hi) {                       // first j with ends[j] > t
      const int mid = (lo + hi) >> 1;
      if (se[mid] > t) hi = mid; else lo = mid + 1;
    }
    const float4* sp = (const float4*)(x + (size_t)(b * TP + lo) * CH);
    v0 = sp[lane * 2];
    v1 = sp[lane * 2 + 1];
  }
  float4* dp = (float4*)(out + ((size_t)b * TMEL + t) * CH);
  dp[lane * 2]     = v0;
  dp[lane * 2 + 1] = v1;
}

// ---------------------------------------------------------------------------
extern "C" void kernel_launch(void* const* d_in, const int* in_sizes, int n_in,
                              void* d_out, int out_size, void* d_ws, size_t ws_size,
                              hipStream_t stream)
{
  const float* x      = (const float*)d_in[0];
  const int*   target = (const int*)d_in[1];
  // d_in[2] = mel_max_length scalar (fixed 2560)
  const float* c1w = (const float*)d_in[3];
  const float* c1b = (const float*)d_in[4];
  const float* g1  = (const float*)d_in[5];
  const float* b1  = (const float*)d_in[6];
  const float* c2w = (const float*)d_in[7];
  const float* c2b = (const float*)d_in[8];
  const float* g2  = (const float*)d_in[9];
  const float* b2  = (const float*)d_in[10];
  const float* lw  = (const float*)d_in[11];
  const float* lb  = (const float*)d_in[12];

  float* out = (float*)d_out;
  const size_t mel_elems = (size_t)NB * TMEL * CH;   // 10,485,760 floats
  float* dur = out + mel_elems;

  // Scratch staged inside the mel-output region (fully overwritten by the
  // final regulate kernel): h1, h2 (2M floats each), wt1, wt2 (196608 each).
  const size_t hsz = (size_t)NB * TP * CH;           // 2,097,152
  const size_t wtsz = 3 * 128 * 512;                 // 196,608
  float* h1  = out;
  float* h2  = out + hsz;
  float* wt1 = out + 2 * hsz;
  float* wt2 = out + 2 * hsz + wtsz;
  int* ends = (int*)d_ws;                            // 32 KB scratch

  transpose_w_kernel<<<(int)(wtsz / 256), 256, 0, stream>>>(c1w, wt1);
  transpose_w_kernel<<<(int)(wtsz / 256), 256, 0, stream>>>(c2w, wt2);

  const dim3 cgrid(TP / 64, NB);
  conv_relu_ln_kernel<<<cgrid, 256, 0, stream>>>(x,  wt1, c1b, g1, b1, h1);
  conv_relu_ln_kernel<<<cgrid, 256, 0, stream>>>(h1, wt2, c2b, g2, b2, h2);
  linear_relu_kernel<<<(NB * TP) / 8, 256, 0, stream>>>(h2, lw, lb, dur);
  cumsum_kernel<<<NB, 512, 0, stream>>>(target, ends);
  regulate_kernel<<<dim3(TMEL / 8, NB), 256, 0, stream>>>(x, ends, out);
}